// HingeLoss_41755672052619
// MI455X (gfx1250) — compile-verified
//
#include <hip/hip_runtime.h>
#include <hip/hip_bf16.h>

// HingeLoss reduction for MI455X (gfx1250).
// Memory-bound: 224 MB compulsory traffic -> ~9.6 us at 23.3 TB/s.
// Strategy: 4 rows/thread with aligned B128 loads, branch-light VALU,
// WMMA-based wave reduction (v_wmma_f32_16x16x4_f32 with ones-B), LDS
// cross-wave combine, per-block partials in d_ws, deterministic final
// single-block reduce into d_out.

#define NC 7

typedef __attribute__((ext_vector_type(2))) float v2f;
typedef __attribute__((ext_vector_type(8))) float v8f;

// Reduce 32 per-lane f32 partials to the wave total using the matrix unit.
// A = [s, 0] per lane (some injective placement into the 16x4 A matrix),
// B = all ones (layout-agnostic). D[m][n] = rowsum_m for every n, so each
// lane's 8 D registers sum to a column sum; adding the xor-16 partner lane's
// column sum gives sum over all 16 rows = the full wave sum in every lane.
// EXEC must be all ones at the call site.
__device__ __forceinline__ float wave_reduce_wmma(float s) {
    v2f a; a[0] = s;    a[1] = 0.0f;
    v2f b; b[0] = 1.0f; b[1] = 1.0f;
    v8f c = {};
    c = __builtin_amdgcn_wmma_f32_16x16x4_f32(
        /*neg_a=*/false, a, /*neg_b=*/false, b,
        /*c_mod=*/(short)0, c, /*reuse_a=*/false, /*reuse_b=*/false);
    float t = ((c[0] + c[1]) + (c[2] + c[3])) + ((c[4] + c[5]) + (c[6] + c[7]));
    t += __shfl_xor(t, 16, 32);
    return t;
}

__global__ void __launch_bounds__(256)
hinge_partial(const float* __restrict__ logit,
              const int* __restrict__ label,
              const long long* __restrict__ repr,
              float* __restrict__ part, int B) {
    const int g  = blockIdx.x * blockDim.x + threadIdx.x;  // group of 4 rows
    const int r0 = g << 2;
    float acc = 0.0f;

    if (r0 + 3 < B) {
        // ---- fast path: fully vectorized, 16B-aligned ----
        const float4* lp = (const float4*)(logit + (size_t)r0 * NC);   // 7 x b128
        float4 L[7];
#pragma unroll
        for (int i = 0; i < 7; ++i) L[i] = lp[i];

        const int4* rp = (const int4*)(repr + (size_t)r0 * 3);          // 6 x b128
        int rep[12];
#pragma unroll
        for (int i = 0; i < 6; ++i) {
            int4 q = rp[i];            // two int64s: low dwords hold 0..6
            rep[2 * i]     = q.x;
            rep[2 * i + 1] = q.z;
        }

        int4 lb = *(const int4*)(label + r0);                           // 1 x b128
        int labs[4] = {lb.x, lb.y, lb.z, lb.w};

        const float* Lf = (const float*)L;
#pragma unroll
        for (int r = 0; r < 4; ++r) {
            const float    t = (labs[r] == 1) ? 1.0f : 0.0f;
            const unsigned m = (1u << rep[3 * r]) | (1u << rep[3 * r + 1]) |
                               (1u << rep[3 * r + 2]);
#pragma unroll
            for (int j = 0; j < NC; ++j) {
                float d = Lf[r * NC + j] - t;
                acc += ((m >> j) & 1u) ? fabsf(d) : fmaxf(d, 0.0f);
            }
        }
    } else if (r0 < B) {
        // ---- scalar tail (at most the last wave) ----
        for (int r = r0; r < B; ++r) {
            const float    t = (label[r] == 1) ? 1.0f : 0.0f;
            const size_t   rb = (size_t)r * 3;
            const unsigned m = (1u << (int)repr[rb]) | (1u << (int)repr[rb + 1]) |
                               (1u << (int)repr[rb + 2]);
            for (int j = 0; j < NC; ++j) {
                float d = logit[(size_t)r * NC + j] - t;
                acc += ((m >> j) & 1u) ? fabsf(d) : fmaxf(d, 0.0f);
            }
        }
    }

    // all 32 lanes of every wave are active here (EXEC all-ones)
    float w = wave_reduce_wmma(acc);

    __shared__ float sdata[8];
    const int lane = threadIdx.x & 31;
    const int wid  = threadIdx.x >> 5;
    if (lane == 0) sdata[wid] = w;
    __syncthreads();
    if (threadIdx.x == 0) {
        float s = 0.0f;
        const int nw = blockDim.x >> 5;
        for (int i = 0; i < nw; ++i) s += sdata[i];
        part[blockIdx.x] = s;
    }
}

__global__ void __launch_bounds__(256)
reduce_final(const float* __restrict__ part, int n, float* __restrict__ out) {
    float acc = 0.0f;
    for (int i = threadIdx.x; i < n; i += blockDim.x) acc += part[i];

    float w = wave_reduce_wmma(acc);  // EXEC all-ones: loop above reconverges

    __shared__ float sdata[8];
    const int lane = threadIdx.x & 31;
    const int wid  = threadIdx.x >> 5;
    if (lane == 0) sdata[wid] = w;
    __syncthreads();
    if (threadIdx.x == 0) {
        float s = 0.0f;
        const int nw = blockDim.x >> 5;
        for (int i = 0; i < nw; ++i) s += sdata[i];
        out[0] = s;
    }
}

extern "C" void kernel_launch(void* const* d_in, const int* in_sizes, int n_in,
                              void* d_out, int out_size, void* d_ws, size_t ws_size,
                              hipStream_t stream) {
    const float*     logit = (const float*)d_in[0];
    const int*       label = (const int*)d_in[1];
    const long long* repr  = (const long long*)d_in[2];

    const int B      = in_sizes[0] / NC;         // 4,000,000
    const int groups = (B + 3) >> 2;             // 4 rows per thread
    const int blocks = (groups + 255) / 256;     // ~3907 blocks

    float* part = (float*)d_ws;                  // blocks * 4 bytes << ws_size

    hinge_partial<<<blocks, 256, 0, stream>>>(logit, label, repr, part, B);
    reduce_final<<<1, 256, 0, stream>>>(part, blocks, (float*)d_out);
}